// TCLT_73641509257879
// MI455X (gfx1250) — compile-verified
//
#include <hip/hip_runtime.h>
#include <math.h>

// ---------------- problem constants (match reference) ----------------
constexpr int Bc    = 32;
constexpr int Nc    = 512;
constexpr int Vc    = 1024;
constexpr int Fc    = Bc * Nc * Nc;      // 8,388,608 = 2^23
constexpr int NN    = Nc * Nc;           // 262,144 = 2^18
constexpr int TLEV  = 64;
constexpr int ILEV  = 64;
constexpr int TROWS = (1 << 21) / TLEV;  // 32768 rows per t-level
constexpr int IROWS = (1 << 20) / ILEV;  // 16384 rows per i-level
constexpr int BDIM  = 256;               // 8 waves (wave32) per block

typedef __attribute__((ext_vector_type(2))) float v2f;
typedef __attribute__((ext_vector_type(8))) float v8f;

// ---------------- wave32 sum-reduction via V_WMMA_F32_16X16X4_F32 ----
// A (16x4 f32): lanes 0-15 hold {K0,K1} for M=lane, lanes 16-31 hold {K2,K3}
// for M=lane-16.  With A = (x,x) per lane and B = 0.5 everywhere:
//   D[M][n] = 0.5 * (x_M + x_M + x_{M+16} + x_{M+16}) = x_M + x_{M+16}
// D layout: lane n<16 holds D[v][n] in VGPR v, lane n+16 holds D[v+8][n].
// Sum of a lane's 8 D regs = sum over half the M rows; one shfl_xor(16)
// completes the full 32-lane sum on every lane.
__device__ __forceinline__ float wave_sum_wmma(float x) {
  v2f a; a.x = x;    a.y = x;
  v2f b; b.x = 0.5f; b.y = 0.5f;
  v8f c = {};
  c = __builtin_amdgcn_wmma_f32_16x16x4_f32(false, a, false, b,
                                            (short)0, c, false, false);
  float s = c[0] + c[1] + c[2] + c[3] + c[4] + c[5] + c[6] + c[7];
  s += __shfl_xor(s, 16, 32);
  return s;
}

__device__ __forceinline__ float wave_max(float x) {
  #pragma unroll
  for (int o = 16; o > 0; o >>= 1) x = fmaxf(x, __shfl_xor(x, o, 32));
  return x;
}

// ---------------- softmax reductions ----------------
__global__ void k_max_part(const float* __restrict__ cost, float* __restrict__ part) {
  int tid = blockIdx.x * BDIM + threadIdx.x;
  int stride = gridDim.x * BDIM;
  float m = -INFINITY;
  for (int i = tid; i < Vc * Vc; i += stride) m = fmaxf(m, cost[i]);
  m = wave_max(m);
  __shared__ float sm[BDIM / 32];
  if ((threadIdx.x & 31) == 0) sm[threadIdx.x >> 5] = m;
  __syncthreads();
  if (threadIdx.x == 0) {
    float r = sm[0];
    #pragma unroll
    for (int i = 1; i < BDIM / 32; ++i) r = fmaxf(r, sm[i]);
    part[blockIdx.x] = r;
  }
}

__global__ void k_max_fin(const float* __restrict__ part, float* __restrict__ scal, int n) {
  float m = (threadIdx.x < n) ? part[threadIdx.x] : -INFINITY;
  m = wave_max(m);
  __shared__ float sm[BDIM / 32];
  if ((threadIdx.x & 31) == 0) sm[threadIdx.x >> 5] = m;
  __syncthreads();
  if (threadIdx.x == 0) {
    float r = sm[0];
    #pragma unroll
    for (int i = 1; i < BDIM / 32; ++i) r = fmaxf(r, sm[i]);
    scal[0] = r;
  }
}

__global__ void k_sum_part(const float* __restrict__ cost, const float* __restrict__ scal,
                           float* __restrict__ part) {
  float gm = scal[0];
  int tid = blockIdx.x * BDIM + threadIdx.x;
  int stride = gridDim.x * BDIM;
  float s = 0.0f;
  for (int i = tid; i < Vc * Vc; i += stride) s += expf(cost[i] - gm);
  s = wave_sum_wmma(s);                      // all lanes active -> EXEC all 1s
  __shared__ float sm[BDIM / 32];
  if ((threadIdx.x & 31) == 0) sm[threadIdx.x >> 5] = s;
  __syncthreads();
  if (threadIdx.x == 0) {
    float r = 0.0f;
    #pragma unroll
    for (int i = 0; i < BDIM / 32; ++i) r += sm[i];
    part[blockIdx.x] = r;
  }
}

__global__ void k_sum_fin(const float* __restrict__ part, float* __restrict__ scal, int n) {
  float s = (threadIdx.x < n) ? part[threadIdx.x] : 0.0f;
  s = wave_sum_wmma(s);
  __shared__ float sm[BDIM / 32];
  if ((threadIdx.x & 31) == 0) sm[threadIdx.x >> 5] = s;
  __syncthreads();
  if (threadIdx.x == 0) {
    float r = 0.0f;
    #pragma unroll
    for (int i = 0; i < BDIM / 32; ++i) r += sm[i];
    scal[1] = r;
  }
}

// ---------------- build mat = softmax gathers ----------------
__global__ void k_build(const float* __restrict__ cost,
                        const int* __restrict__ row_ind, const int* __restrict__ col_ind,
                        const int* __restrict__ r1, const int* __restrict__ c1,
                        const float* __restrict__ scal, float* __restrict__ mat) {
  int f = blockIdx.x * BDIM + threadIdx.x;   // < 2^23
  int b  = f >> 18;
  int ij = f & (NN - 1);
  int i  = ij >> 9;
  int j  = ij & (Nc - 1);
  int ri, cj;
  if (b == 0) {
    ri = row_ind[i];
    cj = col_ind[j];
  } else {
    ri = row_ind[r1[(b - 1) * Nc + i]];
    cj = col_ind[c1[(b - 1) * Nc + j]];
  }
  float gm = scal[0];
  float invs = 1.0f / scal[1];
  mat[f] = expf(cost[ri * Vc + cj] - gm) * invs;
}

// ---------------- t levels: gather phase then scatter phase ----------
__global__ void k_tgather(const int* __restrict__ t_indices, const float* __restrict__ t_costs,
                          const float* __restrict__ pcp, const float* __restrict__ mat,
                          float* __restrict__ vals, int level) {
  int r = blockIdx.x * BDIM + threadIdx.x;       // 0..TROWS-1
  int row = level * TROWS + r;
  const int*   idx = t_indices + (size_t)row * 5;
  const float* c   = t_costs   + (size_t)row * 4;
  float pc = pcp[0];
  float v = -INFINITY;
  #pragma unroll
  for (int k = 0; k < 4; ++k)
    v = fmaxf(v, mat[idx[1 + k]] - c[k] * pc);
  vals[r] = v;
  // warm L2 with next level's index/cost streams (global_prefetch_b8)
  if (level + 1 < TLEV) {
    __builtin_prefetch(t_indices + (size_t)(row + TROWS) * 5, 0, 1);
    __builtin_prefetch(t_costs   + (size_t)(row + TROWS) * 4, 0, 1);
  }
}

__global__ void k_tscatter(const int* __restrict__ t_indices, const float* __restrict__ vals,
                           float* __restrict__ mat, int level) {
  int r = blockIdx.x * BDIM + threadIdx.x;
  int row = level * TROWS + r;
  mat[t_indices[(size_t)row * 5]] = vals[r];
}

// ---------------- i levels ----------------
__global__ void k_igather(const int* __restrict__ i_indices, const float* __restrict__ i_costs,
                          const float* __restrict__ pcp, const float* __restrict__ mat,
                          float* __restrict__ vals, int level) {
  int r = blockIdx.x * BDIM + threadIdx.x;       // 0..IROWS-1
  int row = level * IROWS + r;
  const int*   idx = i_indices + (size_t)row * 9;
  const float* c   = i_costs   + (size_t)row * 4;
  float pc = pcp[0];
  float s0 = mat[idx[1]], s1 = mat[idx[2]], s2 = mat[idx[3]], s3 = mat[idx[4]];
  float s4 = mat[idx[5]], s5 = mat[idx[6]], s6 = mat[idx[7]], s7 = mat[idx[8]];
  float v = fmaxf(s0 + s1, s2 + s3);
  v = fmaxf(v, s4 - c[0] * pc);
  v = fmaxf(v, s5 - c[1] * pc);
  v = fmaxf(v, s6 - c[2] * pc);
  v = fmaxf(v, s7 - c[3] * pc);
  vals[r] = v;
  if (level + 1 < ILEV) {
    __builtin_prefetch(i_indices + (size_t)(row + IROWS) * 9, 0, 1);
    __builtin_prefetch(i_costs   + (size_t)(row + IROWS) * 4, 0, 1);
  }
}

__global__ void k_iscatter(const int* __restrict__ i_indices, const float* __restrict__ vals,
                           float* __restrict__ mat, int level) {
  int r = blockIdx.x * BDIM + threadIdx.x;
  int row = level * IROWS + r;
  mat[i_indices[(size_t)row * 9]] = vals[r];
}

// ---------------- output extraction ----------------
__global__ void k_extract(const float* __restrict__ mat, float* __restrict__ out) {
  int b = threadIdx.x;
  if (b < Bc) out[b] = mat[(size_t)b * NN + (NN - 1)];  // [:, N-1, N-1]
}

// ---------------- host orchestration ----------------
extern "C" void kernel_launch(void* const* d_in, const int* in_sizes, int n_in,
                              void* d_out, int out_size, void* d_ws, size_t ws_size,
                              hipStream_t stream) {
  const float* cost      = (const float*)d_in[0];
  const float* pc        = (const float*)d_in[1];
  const int*   row_ind   = (const int*)d_in[2];
  const int*   col_ind   = (const int*)d_in[3];
  const int*   r1        = (const int*)d_in[4];
  const int*   c1        = (const int*)d_in[5];
  const int*   t_indices = (const int*)d_in[6];
  const float* t_costs   = (const float*)d_in[7];
  const int*   i_indices = (const int*)d_in[8];
  const float* i_costs   = (const float*)d_in[9];
  float* out = (float*)d_out;

  // workspace layout (floats): mat | vals | partials | scalars
  float* w    = (float*)d_ws;
  float* mat  = w;                    // Fc floats (32 MB, L2-resident)
  float* vals = w + Fc;               // TROWS floats (reused for i-levels)
  float* part = vals + TROWS;         // 256 partials
  float* scal = part + 256;           // [0]=gmax, [1]=gsum

  const int RBLOCKS = 256;

  // softmax normalizers
  k_max_part<<<RBLOCKS, BDIM, 0, stream>>>(cost, part);
  k_max_fin <<<1, BDIM, 0, stream>>>(part, scal, RBLOCKS);
  k_sum_part<<<RBLOCKS, BDIM, 0, stream>>>(cost, scal, part);
  k_sum_fin <<<1, BDIM, 0, stream>>>(part, scal, RBLOCKS);

  // mat construction (8M elements)
  k_build<<<Fc / BDIM, BDIM, 0, stream>>>(cost, row_ind, col_ind, r1, c1, scal, mat);

  // 64 t-levels: gather (reads pre-level mat) then scatter (unique dsts)
  for (int L = 0; L < TLEV; ++L) {
    k_tgather <<<TROWS / BDIM, BDIM, 0, stream>>>(t_indices, t_costs, pc, mat, vals, L);
    k_tscatter<<<TROWS / BDIM, BDIM, 0, stream>>>(t_indices, vals, mat, L);
  }

  // 64 i-levels
  for (int L = 0; L < ILEV; ++L) {
    k_igather <<<IROWS / BDIM, BDIM, 0, stream>>>(i_indices, i_costs, pc, mat, vals, L);
    k_iscatter<<<IROWS / BDIM, BDIM, 0, stream>>>(i_indices, vals, mat, L);
  }

  k_extract<<<1, 32, 0, stream>>>(mat, out);
}